// ReluGalois_53317724012560
// MI455X (gfx1250) — compile-verified
//
#include <hip/hip_runtime.h>

// Elementwise Horner polynomial, degree 8 (9 coefficients).
// Memory-bound (AI = 2 FLOP/byte -> ~11.5us at 23.3 TB/s for 268 MB).
// gfx1250 path: b128 non-temporal streaming loads/stores (footprint > L2,
// zero reuse), global_prefetch_b8 four grid-strides ahead, wave32-friendly
// 256-thread blocks, 32-bit indexing so addressing folds into SADDR+VOFFSET.

typedef float v4f __attribute__((ext_vector_type(4)));

#define NCOEF 9
#define TPB   256

static __device__ __forceinline__ v4f splat4(float s) {
    v4f v = {s, s, s, s};
    return v;
}

__global__ __launch_bounds__(TPB) void poly8_v4_kernel(
    const v4f* __restrict__ x,
    const float* __restrict__ coeff,
    v4f* __restrict__ out,
    int n4)
{
    // Broadcast coefficients into registers (uniform address -> scalar loads).
    float c[NCOEF];
#pragma unroll
    for (int k = 0; k < NCOEF; ++k) c[k] = coeff[k];

    const int stride = (int)(gridDim.x * TPB);
    int i = (int)(blockIdx.x * TPB + threadIdx.x);

    // 4 independent b128 NT streams per iteration for memory-level parallelism.
    for (; i + 3 * stride < n4; i += 4 * stride) {
        v4f a0 = __builtin_nontemporal_load(&x[i]);
        v4f a1 = __builtin_nontemporal_load(&x[i + stride]);
        v4f a2 = __builtin_nontemporal_load(&x[i + 2 * stride]);
        v4f a3 = __builtin_nontemporal_load(&x[i + 3 * stride]);

        // Speculative prefetch of next tile (gfx1250: global_prefetch_b8).
        // Harmless if past the end: TH=0 prefetch drops failed translations.
        __builtin_prefetch((const void*)&x[i + 4 * stride], 0, 0);

        v4f r0 = splat4(c[0]);
        v4f r1 = splat4(c[0]);
        v4f r2 = splat4(c[0]);
        v4f r3 = splat4(c[0]);
#pragma unroll
        for (int k = 1; k < NCOEF; ++k) {
            v4f ck = splat4(c[k]);
            r0 = r0 * a0 + ck;
            r1 = r1 * a1 + ck;
            r2 = r2 * a2 + ck;
            r3 = r3 * a3 + ck;
        }

        __builtin_nontemporal_store(r0, &out[i]);
        __builtin_nontemporal_store(r1, &out[i + stride]);
        __builtin_nontemporal_store(r2, &out[i + 2 * stride]);
        __builtin_nontemporal_store(r3, &out[i + 3 * stride]);
    }

    // Remaining full float4s.
    for (; i < n4; i += stride) {
        v4f a = __builtin_nontemporal_load(&x[i]);
        v4f r = splat4(c[0]);
#pragma unroll
        for (int k = 1; k < NCOEF; ++k) r = r * a + splat4(c[k]);
        __builtin_nontemporal_store(r, &out[i]);
    }
}

// Scalar tail for n % 4 != 0 (not hit for 8x2048x2048, kept for robustness).
__global__ void poly8_tail_kernel(const float* __restrict__ x,
                                  const float* __restrict__ coeff,
                                  float* __restrict__ out,
                                  int start, int n)
{
    float c[NCOEF];
#pragma unroll
    for (int k = 0; k < NCOEF; ++k) c[k] = coeff[k];

    for (int i = start + (int)threadIdx.x; i < n; i += (int)blockDim.x) {
        float a = x[i];
        float r = c[0];
#pragma unroll
        for (int k = 1; k < NCOEF; ++k) r = r * a + c[k];
        out[i] = r;
    }
}

extern "C" void kernel_launch(void* const* d_in, const int* in_sizes, int n_in,
                              void* d_out, int out_size, void* d_ws, size_t ws_size,
                              hipStream_t stream) {
    const float* x     = (const float*)d_in[0];
    const float* coeff = (const float*)d_in[1];
    float*       out   = (float*)d_out;

    const int n   = in_sizes[0];          // <= INT_MAX by construction
    const int n4  = n >> 2;
    const int rem = n - (n4 << 2);

    if (n4 > 0) {
        // Grid sized so each thread runs the 4-stream unrolled loop at least
        // twice for this shape: 4096 blocks * 256 thr * 8 f4 = 8M f4 = n4.
        int blocks = 4096;
        int blocks_needed = (n4 / 4 + TPB - 1) / TPB;
        if (blocks_needed < 1) blocks_needed = 1;
        if (blocks_needed < blocks) blocks = blocks_needed;

        poly8_v4_kernel<<<blocks, TPB, 0, stream>>>(
            (const v4f*)x, coeff, (v4f*)out, n4);
    }

    if (rem > 0) {
        poly8_tail_kernel<<<1, 64, 0, stream>>>(x, coeff, out, n4 << 2, n);
    }
}